// PNA_19980187861530
// MI455X (gfx1250) — compile-verified
//
#include <hip/hip_runtime.h>
#include <math.h>

// ---------------- problem constants ----------------
#define NN   3712          // nodes
#define NE   300000        // edges
#define FD   116           // features
#define TT   8             // towers
#define TF   928           // T*F
#define HH   32
#define FO   4             // H/T
#define GG   32
#define CT   8             // col tiles of 16 (116 -> 8, last partial)
#define KS   29            // k-steps of 4 (116/4)
#define AVG_INV 0.349431736f   // 1 / ((ln17+ln18)/2)

typedef __attribute__((ext_vector_type(2))) float v2f;
typedef __attribute__((ext_vector_type(4))) float f4;
typedef __attribute__((ext_vector_type(8))) float v8f;

static __device__ __forceinline__ f4 f4min(f4 a, f4 b){
  f4 r; r[0]=fminf(a[0],b[0]); r[1]=fminf(a[1],b[1]); r[2]=fminf(a[2],b[2]); r[3]=fminf(a[3],b[3]); return r;
}
static __device__ __forceinline__ f4 f4max(f4 a, f4 b){
  f4 r; r[0]=fmaxf(a[0],b[0]); r[1]=fmaxf(a[1],b[1]); r[2]=fmaxf(a[2],b[2]); r[3]=fmaxf(a[3],b[3]); return r;
}

// ---------------- K0: v[t,g], c[t,g] from We slice of Wpre ----------------
__global__ void vc_kernel(const float* __restrict__ Wpre, const float* __restrict__ We_w,
                          const float* __restrict__ We_b, const float* __restrict__ bpre,
                          float* __restrict__ v, float* __restrict__ c){
  int idx = blockIdx.x * blockDim.x + threadIdx.x;
  if (idx >= TT * FD) return;
  int t = idx / FD, g = idx - t * FD;
  const float* Wt = Wpre + ((size_t)t * 3 * FD + 2 * FD) * FD + g;
  float vv = 0.f, cc = 0.f;
  for (int f = 0; f < FD; ++f){ float w = Wt[(size_t)f * FD]; vv += We_w[f] * w; cc += We_b[f] * w; }
  v[idx] = vv; c[idx] = cc + bpre[idx];
}

// ---------------- Kp: pack Wi/Wj into WMMA B-fragment lane order ----------------
// Layout: PB[((t*CT+ct)*KS+ks)*64 + lane*2 + {0,1}]; zero-padded for cols >= 116.
__global__ void pack_b_kernel(const float* __restrict__ Wpre,
                              float* __restrict__ PBi, float* __restrict__ PBj){
  int idx = blockIdx.x * blockDim.x + threadIdx.x;   // (t,ct,ks,lane)
  if (idx >= TT * CT * KS * 32) return;
  int lane = idx & 31;
  int rest = idx >> 5;
  int ks = rest % KS;
  int tc = rest / KS;          // t*CT + ct
  int ct = tc & 7, t = tc >> 3;
  int nl = lane & 15, kb = (lane >> 4) * 2;
  int g = ct * 16 + nl;
  int k = ks * 4 + kb;
  const float* Wt = Wpre + (size_t)t * 3 * FD * FD;
  float bi0 = 0.f, bi1 = 0.f, bj0 = 0.f, bj1 = 0.f;
  if (g < FD){
    bi0 = Wt[(size_t)k * FD + g];
    bi1 = Wt[(size_t)(k + 1) * FD + g];
    bj0 = Wt[(size_t)(FD + k) * FD + g];
    bj1 = Wt[(size_t)(FD + k + 1) * FD + g];
  }
  size_t o = (size_t)idx * 2;
  PBi[o] = bi0; PBi[o + 1] = bi1;
  PBj[o] = bj0; PBj[o + 1] = bj1;
}

// ---------------- K1: ni/nj = x @ Wi[t], x @ Wj[t] via f32 WMMA ----------------
// grid (232 rowtiles, 8 towers), block 256 = 8 waves, wave w owns col tile w.
// Branchless inner loop: b64 A-frag load + 2 pre-packed b64 B-frag loads + 2 WMMAs.
__global__ void gemm_ninj_kernel(const float* __restrict__ x,
                                 const float* __restrict__ PBi, const float* __restrict__ PBj,
                                 float* __restrict__ ni, float* __restrict__ nj){
  const int lane = threadIdx.x & 31;
  const int ct   = threadIdx.x >> 5;
  const int row0 = blockIdx.x * 16;
  const int t    = blockIdx.y;
  const int nl   = lane & 15;          // M for A-frag, N for B/C-frag
  const int kb   = (lane >> 4) * 2;    // K sub-offset 0 or 2
  const size_t fragbase = (size_t)(t * CT + ct) * KS * 64;
  const v2f* pbi = (const v2f*)(PBi + fragbase) + lane;  // +32 v2f per ks
  const v2f* pbj = (const v2f*)(PBj + fragbase) + lane;
  const float* xr = x + (size_t)(row0 + nl) * FD + kb;
  v8f acc_i = {}; v8f acc_j = {};
  #pragma unroll 1
  for (int ks = 0; ks < KS; ++ks){
    v2f a  = *(const v2f*)(xr + ks * 4);
    v2f bi = pbi[ks * 32];
    v2f bj = pbj[ks * 32];
    acc_i = __builtin_amdgcn_wmma_f32_16x16x4_f32(false, a, false, bi, (short)0, acc_i, false, false);
    acc_j = __builtin_amdgcn_wmma_f32_16x16x4_f32(false, a, false, bj, (short)0, acc_j, false, false);
  }
  const int gcol = ct * 16 + nl;
  if (gcol < FD){
    const int mb = (lane >> 4) * 8;
    for (int r = 0; r < 8; ++r){
      size_t o = (size_t)(row0 + mb + r) * TF + t * FD + gcol;
      ni[o] = acc_i[r];
      nj[o] = acc_j[r];
    }
  }
}

// ---------------- CSR build ----------------
__global__ void deg_kernel(const int* __restrict__ edge_index, int* __restrict__ deg){
  int e = blockIdx.x * blockDim.x + threadIdx.x;
  if (e < NE) atomicAdd(&deg[edge_index[NE + e]], 1);
}

__global__ void scan_kernel(const int* __restrict__ deg, int* __restrict__ rowptr, int* __restrict__ cursor){
  __shared__ int lds[256];
  const int tid = threadIdx.x;
  const int CH = (NN + 255) / 256;
  int s = 0;
  for (int j = 0; j < CH; ++j){ int i = tid * CH + j; if (i < NN) s += deg[i]; }
  lds[tid] = s; __syncthreads();
  if (tid == 0){
    int run = 0;
    for (int i = 0; i < 256; ++i){ int tv = lds[i]; lds[i] = run; run += tv; }
    rowptr[NN] = run;
  }
  __syncthreads();
  int run = lds[tid];
  for (int j = 0; j < CH; ++j){
    int i = tid * CH + j;
    if (i < NN){ rowptr[i] = run; cursor[i] = run; run += deg[i]; }
  }
}

__global__ void fill_kernel(const int* __restrict__ edge_index, const float* __restrict__ edge_attr,
                            int* __restrict__ cursor, int* __restrict__ col_src, float* __restrict__ col_ea){
  int e = blockIdx.x * blockDim.x + threadIdx.x;
  if (e >= NE) return;
  int d = edge_index[NE + e];
  int pos = atomicAdd(&cursor[d], 1);
  col_src[pos] = edge_index[e];
  col_ea[pos]  = edge_attr[e];
}

// ---------------- K2: per-node aggregation fused with Wpost + Wlin ----------------
// block = 256 threads, threads 0..231 each own one float4 of the 928 (t,f) columns.
__global__ void agg_kernel(const float* __restrict__ x, const float* __restrict__ ni, const float* __restrict__ nj,
                           const float* __restrict__ vv, const float* __restrict__ cc,
                           const int* __restrict__ rowptr, const int* __restrict__ col_src,
                           const float* __restrict__ col_ea,
                           const float* __restrict__ Wpost, const float* __restrict__ bpost,
                           const float* __restrict__ Wlin, const float* __restrict__ blin,
                           float* __restrict__ out_lin){
  __shared__ int   s_src[64];
  __shared__ float s_ea[64];
  __shared__ float sOut[TT * FO];
  const int n = blockIdx.x, tid = threadIdx.x;
  if (tid < TT * FO) sOut[tid] = bpost[tid];
  const bool act = tid < (TF / 4);
  f4 base = {}, v4 = {};
  if (act){
    base = *(const f4*)(ni + (size_t)n * TF + tid * 4) + *(const f4*)(cc + tid * 4);
    v4   = *(const f4*)(vv + tid * 4);
  }
  const int start = rowptr[n], end = rowptr[n + 1];
  const int deg = end - start;
  f4 sum = {}, sq = {};
  const float BIG = 3.402823466e38f;
  f4 mn = {BIG, BIG, BIG, BIG}, mx = {-BIG, -BIG, -BIG, -BIG};
  __syncthreads();
  for (int eb = start; eb < end; eb += 64){
    int cnt = min(64, end - eb);
    if (tid < cnt){ s_src[tid] = col_src[eb + tid]; s_ea[tid] = col_ea[eb + tid]; }
    __syncthreads();
    if (act){
      for (int j = 0; j < cnt; ++j){
        const int src = s_src[j];
        const float ea = s_ea[j];
        if (j + 1 < cnt)   // hide L2 latency for the next gathered row
          __builtin_prefetch(nj + (size_t)s_src[j + 1] * TF + tid * 4, 0, 3);
        f4 m = base + *(const f4*)(nj + (size_t)src * TF + tid * 4) + ea * v4;
        sum += m; sq += m * m; mn = f4min(mn, m); mx = f4max(mx, m);
      }
    }
    __syncthreads();
  }
  const float dmax = (float)(deg > 0 ? deg : 1);
  const float amp  = logf(dmax + 1.f) * AVG_INV;
  const float iamp = 1.f / amp;
  if (act){
    const float rd = 1.f / dmax;
    f4 mean = sum * rd, mean2 = sq * rd;
    f4 var = mean2 - mean * mean;
    f4 stdv;
    for (int i = 0; i < 4; ++i) stdv[i] = sqrtf(fmaxf(var[i], 0.f) + 1e-5f);
    f4 zero = {};
    f4 mnv = (deg > 0) ? mn : zero;
    f4 mxv = (deg > 0) ? mx : zero;
    const int flat0 = tid * 4;
    const int t  = flat0 / FD;       // F divisible by 4 -> one tower per thread
    const int f0 = flat0 - t * FD;
    const f4* Wp = (const f4*)Wpost; // rows of FOUT=4 floats
    const size_t rbase = (size_t)t * (13 * FD);
    f4 pa = {};
    for (int ci = 0; ci < 4; ++ci){
      const int f = f0 + ci;
      f4 p = x[(size_t)n * FD + f] * Wp[rbase + f];
      const float vals[4] = {mean[ci], mnv[ci], mxv[ci], stdv[ci]};
      for (int k2 = 0; k2 < 4; ++k2){
        const size_t r = rbase + FD + (size_t)k2 * FD + f;
        f4 w = Wp[r] + amp * Wp[r + 4 * FD] + iamp * Wp[r + 8 * FD];
        p += vals[k2] * w;
      }
      pa += p;
    }
    for (int o = 0; o < 4; ++o) atomicAdd(&sOut[t * 4 + o], pa[o]);
  }
  __syncthreads();
  if (tid < HH){
    float y = blin[tid];
    for (int k = 0; k < HH; ++k) y += sOut[k] * Wlin[k * HH + tid];
    out_lin[(size_t)n * HH + tid] = y;
  }
}

// ---------------- BN stats ----------------
__global__ void bn_reduce_kernel(const float* __restrict__ out_lin, float* __restrict__ mu, float* __restrict__ inv){
  __shared__ float s1[256], s2[256];
  const int h = blockIdx.x, tid = threadIdx.x;
  float s = 0.f, q = 0.f;
  for (int n = tid; n < NN; n += 256){ float v = out_lin[(size_t)n * HH + h]; s += v; q += v * v; }
  s1[tid] = s; s2[tid] = q; __syncthreads();
  for (int off = 128; off > 0; off >>= 1){
    if (tid < off){ s1[tid] += s1[tid + off]; s2[tid] += s2[tid + off]; }
    __syncthreads();
  }
  if (tid == 0){
    float m = s1[0] / NN;
    float var = s2[0] / NN - m * m;
    mu[h] = m; inv[h] = rsqrtf(var + 1e-5f);
  }
}

// ---------------- normalize + relu + graph pool ----------------
__global__ void pool_kernel(const float* __restrict__ out_lin, const float* __restrict__ mu,
                            const float* __restrict__ inv, const float* __restrict__ bn_g,
                            const float* __restrict__ bn_b, const int* __restrict__ batch,
                            float* __restrict__ gpool){
  int idx = blockIdx.x * blockDim.x + threadIdx.x;
  if (idx >= NN * HH) return;
  int n = idx >> 5, h = idx & 31;
  float y = (out_lin[idx] - mu[h]) * inv[h] * bn_g[h] + bn_b[h];
  y = fmaxf(y, 0.f);
  atomicAdd(&gpool[batch[n] * HH + h], y);
}

// ---------------- final MLP + log_softmax ----------------
__global__ void mlp_kernel(const float* __restrict__ gpool,
                           const float* __restrict__ W1, const float* __restrict__ b1,
                           const float* __restrict__ W2, const float* __restrict__ b2,
                           const float* __restrict__ W3, const float* __restrict__ b3,
                           float* __restrict__ out){
  int gi = threadIdx.x;
  if (gi >= GG) return;
  const float* r = gpool + gi * HH;
  float z1[16], z2[8], z3[2];
  for (int o = 0; o < 16; ++o){
    float a = b1[o];
    for (int k = 0; k < HH; ++k) a += r[k] * W1[k * 16 + o];
    z1[o] = fmaxf(a, 0.f);
  }
  for (int o = 0; o < 8; ++o){
    float a = b2[o];
    for (int k = 0; k < 16; ++k) a += z1[k] * W2[k * 8 + o];
    z2[o] = fmaxf(a, 0.f);
  }
  for (int o = 0; o < 2; ++o){
    float a = b3[o];
    for (int k = 0; k < 8; ++k) a += z2[k] * W3[k * 2 + o];
    z3[o] = a;
  }
  float m = fmaxf(z3[0], z3[1]);
  float lse = m + logf(expf(z3[0] - m) + expf(z3[1] - m));
  out[gi * 2 + 0] = z3[0] - lse;
  out[gi * 2 + 1] = z3[1] - lse;
}

// ---------------- host launcher ----------------
extern "C" void kernel_launch(void* const* d_in, const int* in_sizes, int n_in,
                              void* d_out, int out_size, void* d_ws, size_t ws_size,
                              hipStream_t stream){
  (void)in_sizes; (void)n_in; (void)out_size; (void)ws_size;
  const float* x         = (const float*)d_in[0];
  const float* edge_attr = (const float*)d_in[1];
  const int*   edge_index= (const int*)  d_in[2];
  const int*   batch     = (const int*)  d_in[3];
  const float* We_w      = (const float*)d_in[4];
  const float* We_b      = (const float*)d_in[5];
  const float* Wpre      = (const float*)d_in[6];
  const float* bpre      = (const float*)d_in[7];
  const float* Wpost     = (const float*)d_in[8];
  const float* bpost     = (const float*)d_in[9];
  const float* Wlin      = (const float*)d_in[10];
  const float* blin      = (const float*)d_in[11];
  const float* bn_g      = (const float*)d_in[12];
  const float* bn_b      = (const float*)d_in[13];
  const float* W1        = (const float*)d_in[14];
  const float* b1        = (const float*)d_in[15];
  const float* W2        = (const float*)d_in[16];
  const float* b2        = (const float*)d_in[17];
  const float* W3        = (const float*)d_in[18];
  const float* b3        = (const float*)d_in[19];

  char* p = (char*)d_ws;
  auto carve = [&](size_t bytes) -> void* {
    void* r = (void*)p;
    p += (bytes + 255) & ~(size_t)255;
    return r;
  };
  const size_t NI = (size_t)NN * TF;
  const size_t PBsz = (size_t)TT * CT * KS * 64;  // packed B fragments per matrix
  float* ni      = (float*)carve(NI * 4);
  float* nj      = (float*)carve(NI * 4);
  float* PBi     = (float*)carve(PBsz * 4);
  float* PBj     = (float*)carve(PBsz * 4);
  float* v       = (float*)carve(TF * 4);
  float* c       = (float*)carve(TF * 4);
  float* col_ea  = (float*)carve((size_t)NE * 4);
  float* out_lin = (float*)carve((size_t)NN * HH * 4);
  float* mu      = (float*)carve(HH * 4);
  float* inv     = (float*)carve(HH * 4);
  float* gpool   = (float*)carve((size_t)GG * HH * 4);
  int*   deg     = (int*)  carve((size_t)NN * 4);
  int*   rowptr  = (int*)  carve((size_t)(NN + 1) * 4);
  int*   cursor  = (int*)  carve((size_t)NN * 4);
  int*   col_src = (int*)  carve((size_t)NE * 4);

  hipMemsetAsync(deg,   0, (size_t)NN * 4,      stream);
  hipMemsetAsync(gpool, 0, (size_t)GG * HH * 4, stream);

  vc_kernel<<<(TT * FD + 255) / 256, 256, 0, stream>>>(Wpre, We_w, We_b, bpre, v, c);
  pack_b_kernel<<<(TT * CT * KS * 32 + 255) / 256, 256, 0, stream>>>(Wpre, PBi, PBj);
  gemm_ninj_kernel<<<dim3(NN / 16, TT), 256, 0, stream>>>(x, PBi, PBj, ni, nj);
  deg_kernel<<<(NE + 255) / 256, 256, 0, stream>>>(edge_index, deg);
  scan_kernel<<<1, 256, 0, stream>>>(deg, rowptr, cursor);
  fill_kernel<<<(NE + 255) / 256, 256, 0, stream>>>(edge_index, edge_attr, cursor, col_src, col_ea);
  agg_kernel<<<NN, 256, 0, stream>>>(x, ni, nj, v, c, rowptr, col_src, col_ea,
                                     Wpost, bpost, Wlin, blin, out_lin);
  bn_reduce_kernel<<<HH, 256, 0, stream>>>(out_lin, mu, inv);
  pool_kernel<<<(NN * HH + 255) / 256, 256, 0, stream>>>(out_lin, mu, inv, bn_g, bn_b, batch, gpool);
  mlp_kernel<<<1, 32, 0, stream>>>(gpool, W1, b1, W2, b2, W3, b3, (float*)d_out);
}